// SparseEdgeEncoder_89515708383729
// MI455X (gfx1250) — compile-verified
//
#include <hip/hip_runtime.h>
#include <hip/hip_bf16.h>
#include <math.h>

// ---------------------------------------------------------------------------
// SparseEdgeEncoder forward for MI455X (gfx1250, wave32).
// Exact algebraic restructuring:
//   (e @ W1)[i*N+j] = feat[i]@W1[:F] + feat[j]@W1[F:]  -> A[i] + B[j]
//   BN1: mean = mean(A)+mean(B)+b1 ; var = Var(A)+Var(B)  (cross term = 0)
//   b1, b2 cancel inside training-mode BatchNorm.
//   Forward hard-gumbel output = [na + g1 > (1-na) + g0], zero diagonal.
// GEMMs: V_WMMA_F32_16X16X4_F32 (full fp32). W1/W2 pre-swizzled into WMMA
// B-fragment order; W1 frags live at LDS offset 0 so every B K-step is one
// ds_load_b64 with a pure immediate offset. GEMM1 K=512 split across all
// 8 waves (two K-segments of 256), partials summed in LDS.
// ---------------------------------------------------------------------------

typedef __attribute__((ext_vector_type(2))) float v2f;
typedef __attribute__((ext_vector_type(8))) float v8f;

#define GRAPHS 128
#define NNODES 30
#define FDIM   512
#define NHID   16
#define NEDGE  (NNODES * NNODES)      // 900
#define MTILES 57                     // ceil(900/16)
#define MPAD   (MTILES * 16)          // 912
#define BN_EPS 1e-5f

// LDS layout (float offsets). Regions reused across phases:
//   W1FR (B-fragment order, at 0 -> DS immediates) -> reused as H1 (912x16)
//   FEAT (32x512, rows 30/31 zero)                 -> reused as Z2 (912x16)
#define OFF_W1   0        // 16384   frag: [half][q<128][lane<32][2]
#define OFF_FEAT 16384    // 16384
#define OFF_W2   32768    // 256     frag: [q<4][lane<32][2]
#define OFF_PAR  33024    // 128: [0]=g1 [1]=be1 [2]=g2 [3]=be2 [4]=W3, [5][0]=b3
#define OFF_STAT 33152    // 128: [0]=mA [1]=mB [2]=vA [3]=vB [4]=rsg1 [5]=m2 [6]=rsg2
#define OFF_A    33280    // 512
#define OFF_B    33792    // 512
#define OFF_P0   34304    // 1024: K-segment-0 partials [half][32][16]
#define OFF_P1   35328    // 1024: K-segment-1 partials
#define OFF_PS   36352    // 256
#define OFF_PSQ  36608    // 256
#define OFF_LG   36864    // 912
#define SMEM_FLOATS 37776
#define SMEM_BYTES  (SMEM_FLOATS * 4)

__device__ __forceinline__ float sigmoidf_(float x) {
    return 1.0f / (1.0f + expf(-x));
}

__global__ __launch_bounds__(256)
void sparse_edge_graph_kernel(const float* __restrict__ nf,
                              const float* __restrict__ W1,
                              const float* __restrict__ g1,
                              const float* __restrict__ be1,
                              const float* __restrict__ W2,
                              const float* __restrict__ g2,
                              const float* __restrict__ be2,
                              const float* __restrict__ W3,
                              const float* __restrict__ b3,
                              const float* __restrict__ U,
                              float* __restrict__ outAdj)
{
    extern __shared__ float sm[];
    const int tid = threadIdx.x;
    const int g   = blockIdx.x;

    // ---- Phase 0: stage inputs into LDS ----
    {
        const float4* fs = (const float4*)(nf + (size_t)g * NNODES * FDIM);
        float4* fd = (float4*)(sm + OFF_FEAT);
        for (int i = tid; i < (NNODES * FDIM) / 4; i += 256) fd[i] = fs[i];
        const float4 z4 = make_float4(0.f, 0.f, 0.f, 0.f);
        for (int i = tid; i < (2 * FDIM) / 4; i += 256) fd[(NNODES * FDIM) / 4 + i] = z4;

        // W1 -> B-fragment order. Each 64-float chunk q (rows 4q..4q+3 of W1)
        // is a local permutation: dest[q*64 + lane*2 + p] = W1[q*64 + (2*hi+p)*16 + n]
        for (int d = tid; d < 2 * FDIM * NHID; d += 256) {
            const int q = d >> 6, r = d & 63;
            const int ln = r >> 1, p = r & 1;
            const int h = ln >> 4, n = ln & 15;
            sm[OFF_W1 + d] = W1[q * 64 + (2 * h + p) * 16 + n];
        }
        // W2 -> same fragment order (4 chunks)
        {
            const int q = tid >> 6, r = tid & 63;
            const int ln = r >> 1, p = r & 1;
            const int h = ln >> 4, n = ln & 15;
            sm[OFF_W2 + tid] = W2[q * 64 + (2 * h + p) * 16 + n];
        }
        if (tid < NHID) {
            sm[OFF_PAR + 0 * 16 + tid] = g1[tid];
            sm[OFF_PAR + 1 * 16 + tid] = be1[tid];
            sm[OFF_PAR + 2 * 16 + tid] = g2[tid];
            sm[OFF_PAR + 3 * 16 + tid] = be2[tid];
            sm[OFF_PAR + 4 * 16 + tid] = W3[tid];
        }
        if (tid == 0) sm[OFF_PAR + 5 * 16] = b3[0];
    }
    __syncthreads();

    const int wid  = tid >> 5;        // wave id (wave32)
    const int lane = tid & 31;
    const int mloc = lane & 15;       // M (A-frag) / N (B,C-frag) position
    const int hi   = lane >> 4;       // selects K pair (A/B) and M+8 (C/D)

    // ---- Phase 1: A = feat @ W1[:512], B = feat @ W1[512:] ----
    // 8 waves: wid&1 = M-tile, (wid>>1)&1 = A/B half, wid>>2 = K segment.
    {
        const int mt   = wid & 1;
        const int half = (wid >> 1) & 1;
        const int ks   = wid >> 2;
        const float* fRow  = sm + OFF_FEAT + (size_t)(mt * 16 + mloc) * FDIM
                           + ks * 256 + 2 * hi;
        const float* wFrag = sm + OFF_W1 + half * (FDIM / 4) * 64
                           + ks * 64 * 64 + lane * 2;
        v8f c = {0.f, 0.f, 0.f, 0.f, 0.f, 0.f, 0.f, 0.f};
        #pragma unroll 8
        for (int q = 0; q < 64; ++q) {
            v2f a = *(const v2f*)(fRow + q * 4);
            v2f b = *(const v2f*)(wFrag + q * 64);
            c = __builtin_amdgcn_wmma_f32_16x16x4_f32(
                    false, a, false, b, (short)0, c, false, false);
        }
        float* dst = sm + (ks == 0 ? OFF_P0 : OFF_P1) + half * 512;
        #pragma unroll
        for (int p = 0; p < 8; ++p) {
            const int m = mt * 16 + p + 8 * hi;   // 0..31, rows 30/31 are zeros
            dst[m * NHID + mloc] = c[p];
        }
    }
    __syncthreads();

    // combine K-segment partials -> A[30][16], B[30][16]
    for (int idx = tid; idx < 2 * NNODES * NHID; idx += 256) {
        const int half = idx >= NNODES * NHID;
        const int rem  = idx - half * NNODES * NHID;
        const float v  = sm[OFF_P0 + half * 512 + rem] + sm[OFF_P1 + half * 512 + rem];
        sm[(half ? OFF_B : OFF_A) + rem] = v;
    }
    __syncthreads();

    // ---- Phase 2: BN1 stats (var = Var(A)+Var(B); b1 cancels) ----
    if (tid < 32) {
        const float* s = sm + (tid < 16 ? OFF_A : OFF_B);
        const int c = tid & 15;
        float su = 0.f, sq = 0.f;
        for (int r = 0; r < NNODES; ++r) {
            const float x = s[r * NHID + c];
            su += x; sq += x * x;
        }
        const float m = su * (1.f / NNODES);
        const float v = sq * (1.f / NNODES) - m * m;
        sm[OFF_STAT + (tid < 16 ? 0 : 1) * 16 + c] = m;
        sm[OFF_STAT + (tid < 16 ? 2 : 3) * 16 + c] = v;
    }
    __syncthreads();
    if (tid < 16) {
        const float v = sm[OFF_STAT + 2 * 16 + tid] + sm[OFF_STAT + 3 * 16 + tid];
        sm[OFF_STAT + 4 * 16 + tid] = rsqrtf(v + BN_EPS) * sm[OFF_PAR + 0 * 16 + tid];
    }
    __syncthreads();

    // ---- Phase 3: h1[r,c] = relu(((A_i-mA)+(B_j-mB))*rsg1 + beta1) ----
    {
        float* h1 = sm + OFF_W1;                      // reuse W1-fragment region
        for (int idx = tid; idx < MPAD * NHID; idx += 256) {
            const int r = idx >> 4, c = idx & 15;
            float val = 0.f;
            if (r < NEDGE) {
                const int i = r / NNODES, j = r - i * NNODES;
                const float x = (sm[OFF_A + i * NHID + c] - sm[OFF_STAT + 0 * 16 + c])
                              + (sm[OFF_B + j * NHID + c] - sm[OFF_STAT + 1 * 16 + c]);
                val = fmaxf(x * sm[OFF_STAT + 4 * 16 + c] + sm[OFF_PAR + 1 * 16 + c], 0.f);
            }
            h1[idx] = val;
        }
    }
    __syncthreads();

    // ---- Phase 4: z2 = h1 @ W2 (57 tiles, 4 chained WMMA each; b2 cancels)
    {
        const float* h1 = sm + OFF_W1;
        float* z2 = sm + OFF_FEAT;                    // reuse feat region
        // W2 B-fragments are tile-invariant: hoist into registers.
        v2f b0 = *(const v2f*)(sm + OFF_W2 + 0 * 64 + lane * 2);
        v2f b1f = *(const v2f*)(sm + OFF_W2 + 1 * 64 + lane * 2);
        v2f b2f = *(const v2f*)(sm + OFF_W2 + 2 * 64 + lane * 2);
        v2f b3f = *(const v2f*)(sm + OFF_W2 + 3 * 64 + lane * 2);
        for (int t = wid; t < MTILES; t += 8) {
            const float* aRow = h1 + (size_t)(t * 16 + mloc) * NHID + 2 * hi;
            v2f a0 = *(const v2f*)(aRow + 0);
            v2f a1 = *(const v2f*)(aRow + 4);
            v2f a2 = *(const v2f*)(aRow + 8);
            v2f a3 = *(const v2f*)(aRow + 12);
            v8f c = {0.f, 0.f, 0.f, 0.f, 0.f, 0.f, 0.f, 0.f};
            c = __builtin_amdgcn_wmma_f32_16x16x4_f32(false, a0, false, b0,  (short)0, c, false, false);
            c = __builtin_amdgcn_wmma_f32_16x16x4_f32(false, a1, false, b1f, (short)0, c, false, false);
            c = __builtin_amdgcn_wmma_f32_16x16x4_f32(false, a2, false, b2f, (short)0, c, false, false);
            c = __builtin_amdgcn_wmma_f32_16x16x4_f32(false, a3, false, b3f, (short)0, c, false, false);
            #pragma unroll
            for (int p = 0; p < 8; ++p) {
                const int m = t * 16 + p + 8 * hi;
                z2[m * NHID + mloc] = c[p];
            }
        }
    }
    __syncthreads();

    // ---- Phase 5: BN2 stats over 900 rows of z2 ----
    {
        const float* z2 = sm + OFF_FEAT;
        const int c = tid & 15, grp = tid >> 4;
        float su = 0.f, sq = 0.f;
        for (int r = grp; r < NEDGE; r += 16) {
            const float x = z2[r * NHID + c];
            su += x; sq += x * x;
        }
        sm[OFF_PS  + grp * 16 + c] = su;
        sm[OFF_PSQ + grp * 16 + c] = sq;
    }
    __syncthreads();
    if (tid < 16) {
        float S = 0.f, SS = 0.f;
        for (int k = 0; k < 16; ++k) {
            S  += sm[OFF_PS  + k * 16 + tid];
            SS += sm[OFF_PSQ + k * 16 + tid];
        }
        const float m = S * (1.f / NEDGE);
        const float v = SS * (1.f / NEDGE) - m * m;
        sm[OFF_STAT + 5 * 16 + tid] = m;
        sm[OFF_STAT + 6 * 16 + tid] = rsqrtf(v + BN_EPS) * sm[OFF_PAR + 2 * 16 + tid];
    }
    __syncthreads();

    // ---- Phase 6: logits[r] = relu(BN2(z2)) @ W3 + b3 ----
    {
        const float* z2 = sm + OFF_FEAT;
        for (int r = tid; r < NEDGE; r += 256) {
            float acc = sm[OFF_PAR + 5 * 16];          // b3
            #pragma unroll
            for (int c = 0; c < NHID; ++c) {
                float h = (z2[r * NHID + c] - sm[OFF_STAT + 5 * 16 + c])
                          * sm[OFF_STAT + 6 * 16 + c] + sm[OFF_PAR + 3 * 16 + c];
                acc += fmaxf(h, 0.f) * sm[OFF_PAR + 4 * 16 + c];
            }
            sm[OFF_LG + r] = acc;
        }
    }
    __syncthreads();

    // ---- Phase 7: sigmoid, symmetrize, straight-through @0.9, hard gumbel ----
    {
        const float* Ug   = U + (size_t)g * NEDGE * 2;
        float*       outg = outAdj + (size_t)g * NEDGE;
        for (int idx = tid; idx < NEDGE; idx += 256) {
            const int i = idx / NNODES, j = idx - i * NNODES;
            const float s  = 0.5f * (sigmoidf_(sm[OFF_LG + idx]) +
                                     sigmoidf_(sm[OFF_LG + j * NNODES + i]));
            const float na = (s > 0.9f) ? s : 0.f;
            const float u0 = Ug[idx * 2 + 0];
            const float u1 = Ug[idx * 2 + 1];
            const float g0 = -logf(-logf(u0));
            const float g1v = -logf(-logf(u1));
            outg[idx] = ((i != j) && ((na + g1v) > (1.f - na + g0))) ? 1.f : 0.f;
        }
    }
}

// labels (as float) + node_features passthrough
__global__ __launch_bounds__(256)
void labels_feat_copy_kernel(const float* __restrict__ nf,
                             const int* __restrict__ labels,
                             float* __restrict__ out)
{
    const int tid = blockIdx.x * blockDim.x + threadIdx.x;
    if (tid < GRAPHS) out[tid] = (float)labels[tid];
    const int n4 = (GRAPHS * NNODES * FDIM) / 4;       // 491520 float4
    const float4* s = (const float4*)nf;
    float4* d = (float4*)(out + GRAPHS + GRAPHS * NEDGE);
    const int stride = gridDim.x * blockDim.x;
    for (int i = tid; i < n4; i += stride) d[i] = s[i];
}

extern "C" void kernel_launch(void* const* d_in, const int* in_sizes, int n_in,
                              void* d_out, int out_size, void* d_ws, size_t ws_size,
                              hipStream_t stream)
{
    const float* nf  = (const float*)d_in[0];
    const float* W1  = (const float*)d_in[1];
    // d_in[2] = b1: cancels inside training-mode BatchNorm, unused
    const float* g1  = (const float*)d_in[3];
    const float* be1 = (const float*)d_in[4];
    const float* W2  = (const float*)d_in[5];
    // d_in[6] = b2: cancels, unused
    const float* g2  = (const float*)d_in[7];
    const float* be2 = (const float*)d_in[8];
    const float* W3  = (const float*)d_in[9];
    const float* b3  = (const float*)d_in[10];
    const float* U   = (const float*)d_in[11];
    const int*   lab = (const int*)d_in[12];
    float* out = (float*)d_out;

    sparse_edge_graph_kernel<<<GRAPHS, 256, SMEM_BYTES, stream>>>(
        nf, W1, g1, be1, W2, g2, be2, W3, b3, U, out + GRAPHS);

    labels_feat_copy_kernel<<<1920, 256, 0, stream>>>(nf, lab, out);
}